// CritiGraph_9448928051400
// MI455X (gfx1250) — compile-verified
//
#include <hip/hip_runtime.h>
#include <hip/hip_bf16.h>
#include <stdint.h>

#define T1 128
#define T2 512
#define TP 8
#define CC 257

// Exact replica of: _, exp = frexp(float(xor+1)); u = 1 - exp/16
// float(z+1) is exact (z+1 <= 65536 < 2^24); frexp exp = biased_exp - 126.
// u = (142 - biased)/16 = 8.875 - biased * 0.0625   (all steps exact)
__device__ __forceinline__ float u_mag(uint32_t z) {
  float f = (float)(z + 1u);
  int biased = (int)(__float_as_uint(f) >> 23);
  return fmaf((float)biased, -0.0625f, 8.875f);
}

// ---------------- Kernel 1: per-(i,j) precompute + j-reduction of w0 -------
// grid = T1 blocks, block = T2 threads (j = tid).
__global__ __launch_bounds__(T2) void k_precompute(
    const int* __restrict__ sta, const int* __restrict__ pos,
    const float* __restrict__ val_n, const float* __restrict__ val_v,
    const unsigned char* __restrict__ mask,
    uint32_t* __restrict__ g_ay, float* __restrict__ g_w2,
    float* __restrict__ g_w3, float* __restrict__ g_w0) {
  __shared__ float red[TP * T2];  // 16 KB
  const int i = blockIdx.x;
  const int j = threadIdx.x;

  const float n  = val_n[i * T2 + j];
  const float v  = val_v[i * T2 + j];
  const float mk = mask[i * T2 + j] ? 1.0f : 0.0f;
  const float n8 = n * 0.125f;

  float d[TP];
  float S = 0.0f;
  uint32_t ayv[TP];
  float syv[TP];
#pragma unroll
  for (int p = 0; p < TP; ++p) {
    const int a = sta[i * TP + p];
    const int b = pos[j * TP + p];
    const uint32_t aa = (uint32_t)(a < 0 ? -a : a);
    const uint32_t ab = (uint32_t)(b < 0 ? -b : b);
    const float sg = ((a < 0) == (b < 0)) ? 1.0f : -1.0f;
    d[p] = sg * u_mag(aa ^ ab) * n;
    S += d[p];
    ayv[p] = ab;
    syv[p] = (b < 0) ? -1.0f : 1.0f;
  }

  if (i == 0) {  // |pos| is i-independent; write once
#pragma unroll
    for (int p = 0; p < TP; ++p) g_ay[p * T2 + j] = ayv[p];
  }
  g_w3[i * T2 + j] = mk * n8 * n8;

#pragma unroll
  for (int p = 0; p < TP; ++p) {
    const float A = (S - d[p]) * 0.125f - v;          // c-independent part
    g_w2[(i * TP + p) * T2 + j] = 2.0f * mk * A * n8 * syv[p];  // sy folded in
    red[p * T2 + j] = mk * A * A;
  }

  // deterministic tree reduction over j for w0[i,p]
  for (int s = T2 / 2; s > 0; s >>= 1) {
    __syncthreads();
    if (j < s) {
#pragma unroll
      for (int p = 0; p < TP; ++p) red[p * T2 + j] += red[p * T2 + j + s];
    }
  }
  __syncthreads();
  if (j < TP) g_w0[i * TP + j] = red[j * T2];
}

// ---------------- Kernel 2: main j-loop over LDS-staged data ---------------
// grid = (T1, ceil(C/32)), block = 256 (wave w -> p = w, lane -> c)
#if defined(__gfx1250__) && __has_builtin(__builtin_amdgcn_global_load_async_to_lds_b128)
#define USE_ASYNC_LDS 1
typedef int v4i_t __attribute__((ext_vector_type(4)));
typedef __attribute__((address_space(1))) v4i_t* gv4p;
typedef __attribute__((address_space(3))) v4i_t* lv4p;
#endif

__global__ __launch_bounds__(256) void k_loss(
    const int* __restrict__ cnc, const float* __restrict__ lth,
    const uint32_t* __restrict__ g_ay, const float* __restrict__ g_w2,
    const float* __restrict__ g_w3, const float* __restrict__ g_w0,
    float* __restrict__ out) {
  __shared__ __align__(16) uint32_t s_ay[TP * T2];  // 16 KB
  __shared__ __align__(16) float    s_w2[TP * T2];  // 16 KB
  __shared__ __align__(16) float    s_w3[T2];       //  2 KB

  const int i   = blockIdx.x;
  const int tid = threadIdx.x;

#if USE_ASYNC_LDS
  {
    gv4p ga  = (gv4p)(uintptr_t)g_ay;
    lv4p la  = (lv4p)s_ay;
    for (int k = tid; k < (TP * T2 * 4) / 16; k += 256)
      __builtin_amdgcn_global_load_async_to_lds_b128(ga + k, la + k, 0, 0);
    gv4p gw2 = (gv4p)(uintptr_t)(g_w2 + (size_t)i * TP * T2);
    lv4p lw2 = (lv4p)s_w2;
    for (int k = tid; k < (TP * T2 * 4) / 16; k += 256)
      __builtin_amdgcn_global_load_async_to_lds_b128(gw2 + k, lw2 + k, 0, 0);
    gv4p gw3 = (gv4p)(uintptr_t)(g_w3 + (size_t)i * T2);
    lv4p lw3 = (lv4p)s_w3;
    for (int k = tid; k < (T2 * 4) / 16; k += 256)
      __builtin_amdgcn_global_load_async_to_lds_b128(gw3 + k, lw3 + k, 0, 0);
  }
#if __has_builtin(__builtin_amdgcn_s_wait_asynccnt)
  __builtin_amdgcn_s_wait_asynccnt(0);
#else
  asm volatile("s_wait_asynccnt 0" ::: "memory");
#endif
  __syncthreads();
#else
  {
    const uint4*  ga4  = (const uint4*)g_ay;
    uint4*        la4  = (uint4*)s_ay;
    for (int k = tid; k < (TP * T2) / 4; k += 256) la4[k] = ga4[k];
    const float4* gw24 = (const float4*)(g_w2 + (size_t)i * TP * T2);
    float4*       lw24 = (float4*)s_w2;
    for (int k = tid; k < (TP * T2) / 4; k += 256) lw24[k] = gw24[k];
    const float4* gw34 = (const float4*)(g_w3 + (size_t)i * T2);
    float4*       lw34 = (float4*)s_w3;
    for (int k = tid; k < T2 / 4; k += 256) lw34[k] = gw34[k];
  }
  __syncthreads();
#endif

  const int p    = tid >> 5;
  const int lane = tid & 31;
  const int c    = blockIdx.y * 32 + lane;

  int x = 0;
  if (c < CC) x = cnc[((size_t)i * CC + c) * TP + p];
  const uint32_t ax  = (uint32_t)(x < 0 ? -x : x);
  const float    sgx = (x < 0) ? -1.0f : 1.0f;

  const uint4*  ay4 = (const uint4*)(&s_ay[p * T2]);  // wave-uniform addresses
  const float4* w24 = (const float4*)(&s_w2[p * T2]);
  const float4* w34 = (const float4*)(s_w3);

  float acc2 = 0.0f, acc3 = 0.0f;
#pragma unroll 4
  for (int q = 0; q < T2 / 4; ++q) {
    const uint4  za = ay4[q];
    const float4 w2 = w24[q];
    const float4 w3 = w34[q];
    float u;
    u = u_mag(ax ^ za.x); acc2 = fmaf(w2.x, u, acc2); acc3 = fmaf(w3.x, u * u, acc3);
    u = u_mag(ax ^ za.y); acc2 = fmaf(w2.y, u, acc2); acc3 = fmaf(w3.y, u * u, acc3);
    u = u_mag(ax ^ za.z); acc2 = fmaf(w2.z, u, acc2); acc3 = fmaf(w3.z, u * u, acc3);
    u = u_mag(ax ^ za.w); acc2 = fmaf(w2.w, u, acc2); acc3 = fmaf(w3.w, u * u, acc3);
  }

  const float loss = (g_w0[i * TP + p] + sgx * acc2 + acc3) / lth[i];
  if (c < CC) out[((size_t)i * CC + c) * TP + p] = loss;
}

// ---------------------------------------------------------------------------
extern "C" void kernel_launch(void* const* d_in, const int* in_sizes, int n_in,
                              void* d_out, int out_size, void* d_ws, size_t ws_size,
                              hipStream_t stream) {
  const int*           sta = (const int*)d_in[0];
  const int*           pos = (const int*)d_in[1];
  const int*           cnc = (const int*)d_in[2];
  const float*         vn  = (const float*)d_in[3];
  const float*         vv  = (const float*)d_in[4];
  const unsigned char* mk  = (const unsigned char*)d_in[5];
  const float*         lth = (const float*)d_in[6];
  float*               out = (float*)d_out;

  char* ws = (char*)d_ws;
  uint32_t* g_ay = (uint32_t*)ws;                                  // TP*T2 u32   (16 KB)
  float* g_w2 = (float*)(ws + 16384);                              // T1*TP*T2 f32 (2 MB)
  float* g_w3 = (float*)(ws + 16384 + 2097152);                    // T1*T2 f32  (256 KB)
  float* g_w0 = (float*)(ws + 16384 + 2097152 + 262144);           // T1*TP f32  (4 KB)

  k_precompute<<<T1, T2, 0, stream>>>(sta, pos, vn, vv, mk, g_ay, g_w2, g_w3, g_w0);
  dim3 grid(T1, (CC + 31) / 32);
  k_loss<<<grid, 256, 0, stream>>>(cnc, lth, g_ay, g_w2, g_w3, g_w0, out);
}